// TPP_47897475285383
// MI455X (gfx1250) — compile-verified
//
#include <hip/hip_runtime.h>
#include <math.h>

// CDNA5 / gfx1250, wave32. FP32 WMMA path: V_WMMA_F32_16X16X4_F32.
typedef float v2f __attribute__((ext_vector_type(2)));
typedef float v8f __attribute__((ext_vector_type(8)));

#define S_IN   128
#define S_OUTD 128
#define V_IN   64
#define V_OUTD 64
#define HID    16
#define CAT    153   // 128 + 9 + 16

__device__ __forceinline__ v8f zero8() {
    v8f r = {0.f, 0.f, 0.f, 0.f, 0.f, 0.f, 0.f, 0.f};
    return r;
}

// D = A(16x4 f32) * B(4x16 f32) + C(16x16 f32)
__device__ __forceinline__ v8f wmma4(v2f a, v2f b, v8f c) {
    return __builtin_amdgcn_wmma_f32_16x16x4_f32(
        /*neg_a=*/false, a, /*neg_b=*/false, b,
        /*c_mod=*/(short)0, c, /*reuse_a=*/false, /*reuse_b=*/false);
}

__device__ __forceinline__ float sigm(float x) { return 1.0f / (1.0f + __expf(-x)); }
__device__ __forceinline__ float silu(float x) { return x * sigm(x); }

// Per-wave LDS scratch (floats)
#define SBUF_OFF 0       // 16*128  : pre-activation s_out rows
#define CX_OFF   2048    // 16*32   : cols 0..8 scal, 9..24 norm, 25..27 zero pad
#define Z_OFF    2560    // 3*16*16 : z tiles (f, row, hid)
#define WAVE_LDS 3328    // floats per wave; 4 waves -> 53,248 B static LDS

__global__ __launch_bounds__(128) void tpp_kernel(
    const float* __restrict__ s, const float* __restrict__ v,
    const float* __restrict__ frames,
    const float* __restrict__ Wd,  const float* __restrict__ Wds,
    const float* __restrict__ bds, const float* __restrict__ Wu,
    const float* __restrict__ Wso, const float* __restrict__ bso,
    const float* __restrict__ Wgt, const float* __restrict__ bgt,
    float* __restrict__ out_s, float* __restrict__ out_v, int N)
{
    __shared__ float lds[4 * WAVE_LDS];
    const int lane  = threadIdx.x & 31;
    const int wave  = threadIdx.x >> 5;
    float* wl = lds + wave * WAVE_LDS;

    const int m     = lane & 15;   // A-row / C-column index
    const int h     = lane >> 4;   // half-wave
    const int hb    = h * 2;       // K sub-offset within 4-wide K window
    const int col16 = m;

    const int tile = blockIdx.x * 4 + wave;   // one 16-point tile per wave
    const int row0 = tile * 16;
    if (row0 >= N) return;                    // wave-uniform exit
    const int rowm = (row0 + m < N) ? (row0 + m) : (N - 1);

    const float* vrow = v + (size_t)rowm * (V_IN * 3);

    // ---- Stage 1: vs = silu(v_t @ W_down_s + b); scal = frames @ vs (VALU) ----
    float part[9];
    #pragma unroll
    for (int i = 0; i < 9; ++i) part[i] = 0.f;
    const int jbase = 32 * h;
    for (int j = jbase; j < jbase + 32; ++j) {
        const float a0 = vrow[j * 3 + 0], a1 = vrow[j * 3 + 1], a2 = vrow[j * 3 + 2];
        const float w0 = Wds[j * 3 + 0],  w1 = Wds[j * 3 + 1],  w2 = Wds[j * 3 + 2];
        part[0] += a0 * w0; part[1] += a0 * w1; part[2] += a0 * w2;
        part[3] += a1 * w0; part[4] += a1 * w1; part[5] += a1 * w2;
        part[6] += a2 * w0; part[7] += a2 * w1; part[8] += a2 * w2;
    }
    float vsv[9];
    #pragma unroll
    for (int i = 0; i < 9; ++i) {
        float t = part[i] + __shfl_xor(part[i], 16, 32);  // combine half-wave partials
        t += bds[i % 3];
        vsv[i] = silu(t);
    }
    {
        const float* fr = frames + (size_t)rowm * 9;
        if (h == 0) {
            #pragma unroll
            for (int fo = 0; fo < 3; ++fo)
                #pragma unroll
                for (int d = 0; d < 3; ++d)
                    wl[CX_OFF + m * 32 + fo * 3 + d] =
                        fr[fo * 3 + 0] * vsv[0 * 3 + d] +
                        fr[fo * 3 + 1] * vsv[1 * 3 + d] +
                        fr[fo * 3 + 2] * vsv[2 * 3 + d];
            wl[CX_OFF + m * 32 + 25] = 0.f;   // K padding 153..155
            wl[CX_OFF + m * 32 + 26] = 0.f;
            wl[CX_OFF + m * 32 + 27] = 0.f;
        }
    }

    // ---- Stage 2: z_f[16x16] = V_f[16x64] @ W_down[64x16]  (3 x 16 WMMA) ----
    v8f zacc[3];
    #pragma unroll
    for (int f = 0; f < 3; ++f) {
        v8f acc = zero8();
        #pragma unroll
        for (int kk = 0; kk < 16; ++kk) {
            const int k0 = 4 * kk + hb;
            v2f a = { vrow[k0 * 3 + f], vrow[(k0 + 1) * 3 + f] };
            v2f b = { Wd[k0 * HID + col16], Wd[(k0 + 1) * HID + col16] };
            acc = wmma4(a, b, acc);
        }
        zacc[f] = acc;
    }
    // norms -> cx cols 9..24; z tiles -> z scratch (A-layout source for v_up)
    #pragma unroll
    for (int g = 0; g < 8; ++g) {
        const int mm = g + 8 * h;
        const float z0 = zacc[0][g], z1 = zacc[1][g], z2 = zacc[2][g];
        wl[CX_OFF + mm * 32 + 9 + col16] = sqrtf(z0 * z0 + z1 * z1 + z2 * z2);
        wl[Z_OFF + 0 * 256 + mm * 16 + col16] = z0;
        wl[Z_OFF + 1 * 256 + mm * 16 + col16] = z1;
        wl[Z_OFF + 2 * 256 + mm * 16 + col16] = z2;
    }

    // ---- Stage 3: S[16x128] = concat[16x153] @ W_sout  (8 x 39 WMMA, K padded) ----
    v8f sacc[8];
    #pragma unroll
    for (int nt = 0; nt < 8; ++nt) sacc[nt] = zero8();
    const float* srow = s + (size_t)rowm * S_IN;
    for (int kk = 0; kk < 38; ++kk) {
        const int k0 = 4 * kk + hb;
        v2f a;
        if (kk < 32) {                       // cols 0..127 straight from global s
            a.x = srow[k0];
            a.y = srow[k0 + 1];
        } else {                             // cols 128..151 from LDS cx
            const int c0 = k0 - 128;
            a.x = wl[CX_OFF + m * 32 + c0];
            a.y = wl[CX_OFF + m * 32 + c0 + 1];
        }
        #pragma unroll
        for (int nt = 0; nt < 8; ++nt) {
            const int colg = nt * 16 + col16;
            v2f b = { Wso[(size_t)k0 * S_OUTD + colg],
                      Wso[(size_t)(k0 + 1) * S_OUTD + colg] };
            sacc[nt] = wmma4(a, b, sacc[nt]);
        }
    }
    {   // kk = 38 : k = 152..155, rows >= 153 are zero
        const int k0 = 152 + hb;
        v2f a = { wl[CX_OFF + m * 32 + 24 + hb], wl[CX_OFF + m * 32 + 25 + hb] };
        #pragma unroll
        for (int nt = 0; nt < 8; ++nt) {
            const int colg = nt * 16 + col16;
            v2f b = { (k0     < CAT) ? Wso[(size_t)k0 * S_OUTD + colg]       : 0.f,
                      (k0 + 1 < CAT) ? Wso[(size_t)(k0 + 1) * S_OUTD + colg] : 0.f };
            sacc[nt] = wmma4(a, b, sacc[nt]);
        }
    }
    // bias; pre-activation S -> LDS (gate GEMM input); silu(S) -> global out
    #pragma unroll
    for (int nt = 0; nt < 8; ++nt) {
        const int colg = nt * 16 + col16;
        const float bias = bso[colg];
        #pragma unroll
        for (int g = 0; g < 8; ++g) {
            const int mm = g + 8 * h;
            const float Sv = sacc[nt][g] + bias;
            wl[SBUF_OFF + mm * S_IN + colg] = Sv;
            const int rr = row0 + mm;
            if (rr < N) out_s[(size_t)rr * S_OUTD + colg] = silu(Sv);
        }
    }

    // ---- Stage 4: gate[16x64] = sigmoid(S[16x128] @ W_gate + b)  (4 x 32 WMMA) ----
    v8f gacc[4];
    #pragma unroll
    for (int nt = 0; nt < 4; ++nt) gacc[nt] = zero8();
    for (int kk = 0; kk < 32; ++kk) {
        const int k0 = 4 * kk + hb;
        v2f a = { wl[SBUF_OFF + m * S_IN + k0], wl[SBUF_OFF + m * S_IN + k0 + 1] };
        #pragma unroll
        for (int nt = 0; nt < 4; ++nt) {
            const int colg = nt * 16 + col16;
            v2f b = { Wgt[k0 * V_OUTD + colg], Wgt[(k0 + 1) * V_OUTD + colg] };
            gacc[nt] = wmma4(a, b, gacc[nt]);
        }
    }
    #pragma unroll
    for (int nt = 0; nt < 4; ++nt) {
        const float bias = bgt[nt * 16 + col16];
        #pragma unroll
        for (int g = 0; g < 8; ++g) gacc[nt][g] = sigm(gacc[nt][g] + bias);
    }

    // ---- Stage 5: vup_f[16x64] = z_f @ W_up  (3 x 4 x 4 WMMA) ; gate & store ----
    v8f vacc[3][4];
    #pragma unroll
    for (int f = 0; f < 3; ++f)
        #pragma unroll
        for (int nt = 0; nt < 4; ++nt) vacc[f][nt] = zero8();
    #pragma unroll
    for (int f = 0; f < 3; ++f) {
        #pragma unroll
        for (int kk = 0; kk < 4; ++kk) {
            const int k0 = 4 * kk + hb;
            v2f a = { wl[Z_OFF + f * 256 + m * 16 + k0],
                      wl[Z_OFF + f * 256 + m * 16 + k0 + 1] };
            #pragma unroll
            for (int nt = 0; nt < 4; ++nt) {
                const int colg = nt * 16 + col16;
                v2f b = { Wu[k0 * V_OUTD + colg], Wu[(k0 + 1) * V_OUTD + colg] };
                vacc[f][nt] = wmma4(a, b, vacc[f][nt]);
            }
        }
    }
    #pragma unroll
    for (int nt = 0; nt < 4; ++nt) {
        const int colg = nt * 16 + col16;
        #pragma unroll
        for (int g = 0; g < 8; ++g) {
            const int mm = g + 8 * h;
            const int rr = row0 + mm;
            if (rr < N) {
                const float gt = gacc[nt][g];
                const size_t o = ((size_t)rr * V_OUTD + colg) * 3;
                out_v[o + 0] = vacc[0][nt][g] * gt;
                out_v[o + 1] = vacc[1][nt][g] * gt;
                out_v[o + 2] = vacc[2][nt][g] * gt;
            }
        }
    }
}

extern "C" void kernel_launch(void* const* d_in, const int* in_sizes, int n_in,
                              void* d_out, int out_size, void* d_ws, size_t ws_size,
                              hipStream_t stream) {
    const float* s   = (const float*)d_in[0];
    const float* v   = (const float*)d_in[1];
    const float* fr  = (const float*)d_in[2];
    const float* Wd  = (const float*)d_in[3];
    const float* Wds = (const float*)d_in[4];
    const float* bds = (const float*)d_in[5];
    const float* Wu  = (const float*)d_in[6];
    const float* Wso = (const float*)d_in[7];
    const float* bso = (const float*)d_in[8];
    const float* Wgt = (const float*)d_in[9];
    const float* bgt = (const float*)d_in[10];

    const int N = in_sizes[0] / S_IN;         // 400000
    float* out_s = (float*)d_out;             // [N,128] then [N,64,3]
    float* out_v = out_s + (size_t)N * S_OUTD;

    const int tiles  = (N + 15) / 16;         // one wave per 16-point tile
    const int blocks = (tiles + 3) / 4;       // 4 waves (128 threads) per block
    tpp_kernel<<<blocks, 128, 0, stream>>>(s, v, fr, Wd, Wds, bds, Wu,
                                           Wso, bso, Wgt, bgt, out_s, out_v, N);
}